// Model_11278584119598
// MI455X (gfx1250) — compile-verified
//
#include <hip/hip_runtime.h>

// ---------------------------------------------------------------------------
// CDNA5 fp32 WMMA: D(16x16) = A(16x4) x B(4x16) + C
// A,B = 2 VGPRs (v2f), C/D = 8 VGPRs (v8f), wave32.
// Lane layout (ISA 7.12.2): m = lane%16, hi = lane/16;
//   A: lane holds A[m][k0+2*hi], A[m][k0+2*hi+1]   (one b64 pair)
//   B: lane holds B[k0+2*hi][n], B[k0+2*hi+1][n]   (one b64 pair if packed)
//   D: VGPR v -> row v (lanes 0-15) / row v+8 (lanes 16-31), col = m
// ---------------------------------------------------------------------------
typedef __attribute__((ext_vector_type(2))) float v2f;
typedef __attribute__((ext_vector_type(8))) float v8f;

__device__ inline v8f wmma_f32_16x16x4(v2f a, v2f b, v8f c) {
  return __builtin_amdgcn_wmma_f32_16x16x4_f32(
      false, a, false, b, (short)0, c, false, false);
}

__device__ inline int imin(int a, int b) { return a < b ? a : b; }
__device__ inline int imax(int a, int b) { return a > b ? a : b; }

// ---------------------------------------------------------------------------
// Resize-chain step descriptors (PyTorch-faithful semantics, fp32 math).
// mode: 0 = nearest (scale = in/out), 1 = linear, 2 = cubic (A = -0.75)
// ---------------------------------------------------------------------------
struct RStep { int out; int mode; int ac; };

__device__ const RStep kStepsX[11] = {
    {16,0,0},{32,0,0},{20,0,0},{80,0,0},
    {16,1,0},{32,1,0},{24,1,1},{72,1,1},
    {108,0,0},{129,1,0},{103,1,1}};

__device__ const RStep kStepsYR[23] = {
    {16,0,0},{32,0,0},{20,0,0},{80,0,0},{16,0,0},{32,0,0},
    {16,1,0},{32,1,0},{20,1,0},{80,1,0},{16,1,1},{32,1,1},
    {16,2,0},{32,2,0},{20,2,0},{80,2,0},{16,2,1},{32,2,1},
    {51,0,0},{102,1,0},{51,1,1},{40,2,0},{44,2,1}};

__device__ const RStep kStepsYC[23] = {
    {16,0,0},{32,0,0},{20,0,0},{80,0,0},{24,0,0},{72,0,0},
    {16,1,0},{32,1,0},{20,1,0},{80,1,0},{24,1,1},{72,1,1},
    {16,2,0},{32,2,0},{20,2,0},{80,2,0},{24,2,1},{72,2,1},
    {144,0,0},{172,1,0},{68,1,1},{61,2,0},{30,2,1}};

#define BPITCH 192   // max intermediate length: 172 (y cols), 129 (x)
#define BROWS  64

// One output sample of a 1-D resize applied to row `base` of src (pitch BPITCH).
__device__ float resize_sample(const float* src, int base, int n, int outN,
                               int j, int mode, int ac) {
  if (mode == 0) { // nearest: idx = clamp(floor(j * f32(n/out)), 0, n-1)
    float s = (float)((double)n / (double)outN);
    int i = (int)floorf((float)j * s);
    i = imax(0, imin(i, n - 1));
    return src[base + i];
  }
  float srcf;
  if (ac) {
    double s = (outN > 1) ? (double)(n - 1) / (double)(outN - 1) : 0.0;
    srcf = (float)j * (float)s;
  } else {
    srcf = ((float)j + 0.5f) * (float)((double)n / (double)outN) - 0.5f;
  }
  if (mode == 1) { // linear: clamp src >= 0
    float sc = fmaxf(srcf, 0.0f);
    int i0 = imin(imax((int)floorf(sc), 0), n - 1);
    int i1 = imin(i0 + 1, n - 1);
    float t = sc - (float)i0;
    return src[base + i0] * (1.0f - t) + src[base + i1] * t;
  }
  // cubic (no src clamp), 4 taps with clamped indices
  const float A = -0.75f;
  int i = (int)floorf(srcf);
  float t = srcf - (float)i;
  float u0 = t + 1.0f, u1 = t, u2 = 1.0f - t, u3 = 2.0f - t;
  float w0 = ((A * u0 - 5.0f * A) * u0 + 8.0f * A) * u0 - 4.0f * A;
  float w1 = ((A + 2.0f) * u1 - (A + 3.0f)) * u1 * u1 + 1.0f;
  float w2 = ((A + 2.0f) * u2 - (A + 3.0f)) * u2 * u2 + 1.0f;
  float w3 = ((A * u3 - 5.0f * A) * u3 + 8.0f * A) * u3 - 4.0f * A;
  int p0 = imin(imax(i - 1, 0), n - 1);
  int p1 = imin(imax(i    , 0), n - 1);
  int p2 = imin(imax(i + 1, 0), n - 1);
  int p3 = imin(imax(i + 2, 0), n - 1);
  return w0 * src[base + p0] + w1 * src[base + p1] +
         w2 * src[base + p2] + w3 * src[base + p3];
}

// ---------------------------------------------------------------------------
// Builder: push identity through each chain, then emit PAIR-PACKED, PADDED
// operand matrices so the GEMM k-loops use unconditional b64 loads:
//   chain 0 -> MxP : float2[32][112]  MxP[k2][c] = {Mx[2k2][c], Mx[2k2+1][c]}, c>=103 -> 0
//   chain 1 -> GrP : float2[16][48]   GrP[k2][m] = {Gr[2k2][m], Gr[2k2+1][m]}, m>=44  -> 0
//   chain 2 -> GcP : float2[16][32]   GcP[k2][c] = {Gc[2k2][c], Gc[2k2+1][c]}, c>=30  -> 0
// ---------------------------------------------------------------------------
__global__ void build_mats_kernel(float* __restrict__ scratch,
                                  float* __restrict__ MxP,
                                  float* __restrict__ GrP,
                                  float* __restrict__ GcP) {
  const int chain = blockIdx.x;
  int rows, n0, nsteps;
  const RStep* steps;
  if (chain == 0)      { rows = 64; n0 = 64; nsteps = 11; steps = kStepsX;  }
  else if (chain == 1) { rows = 32; n0 = 32; nsteps = 23; steps = kStepsYR; }
  else                 { rows = 32; n0 = 32; nsteps = 23; steps = kStepsYC; }

  float* buf0 = scratch + (size_t)chain * 2u * BROWS * BPITCH;
  float* buf1 = buf0 + BROWS * BPITCH;

  for (int t = threadIdx.x; t < rows * n0; t += blockDim.x) {
    int r = t / n0, c = t - r * n0;
    buf0[r * BPITCH + c] = (r == c) ? 1.0f : 0.0f;
  }
  __threadfence_block();
  __syncthreads();

  float* cur = buf0; float* nxt = buf1;
  int n = n0;
  for (int s = 0; s < nsteps; ++s) {
    const int outN = steps[s].out, mode = steps[s].mode, ac = steps[s].ac;
    for (int t = threadIdx.x; t < rows * outN; t += blockDim.x) {
      int r = t / outN, j = t - r * outN;
      nxt[r * BPITCH + j] = resize_sample(cur, r * BPITCH, n, outN, j, mode, ac);
    }
    __threadfence_block();
    __syncthreads();
    float* tmp = cur; cur = nxt; nxt = tmp;
    n = outN;
  }

  // Pair-pack + pad. cur[k][j] with k = input index, j = output index.
  if (chain == 0) {
    for (int t = threadIdx.x; t < 32 * 112; t += blockDim.x) {
      int k2 = t / 112, c = t - k2 * 112;
      float v0 = (c < 103) ? cur[(2 * k2)     * BPITCH + c] : 0.0f;
      float v1 = (c < 103) ? cur[(2 * k2 + 1) * BPITCH + c] : 0.0f;
      MxP[2 * t] = v0; MxP[2 * t + 1] = v1;
    }
  } else if (chain == 1) {
    for (int t = threadIdx.x; t < 16 * 48; t += blockDim.x) {
      int k2 = t / 48, m = t - k2 * 48;
      float v0 = (m < 44) ? cur[(2 * k2)     * BPITCH + m] : 0.0f;
      float v1 = (m < 44) ? cur[(2 * k2 + 1) * BPITCH + m] : 0.0f;
      GrP[2 * t] = v0; GrP[2 * t + 1] = v1;
    }
  } else {
    for (int t = threadIdx.x; t < 16 * 32; t += blockDim.x) {
      int k2 = t / 32, c = t - k2 * 32;
      float v0 = (c < 30) ? cur[(2 * k2)     * BPITCH + c] : 0.0f;
      float v1 = (c < 30) ? cur[(2 * k2 + 1) * BPITCH + c] : 0.0f;
      GcP[2 * t] = v0; GcP[2 * t + 1] = v1;
    }
  }
}

// ---------------------------------------------------------------------------
// x: out(2048x103) = X(2048x64) @ Mx(64x103, padded to 112 cols).
// One wave per 16x16 tile; K=64 -> 16 WMMA steps; all loads unconditional b64.
// ---------------------------------------------------------------------------
__global__ void x_resize_gemm(const float* __restrict__ X,
                              const float* __restrict__ MxP,
                              float* __restrict__ outX) {
  const int tile = blockIdx.x;
  const int rt = tile / 7, ct = tile - rt * 7;
  const int lane = threadIdx.x;
  const int m = lane & 15, hi = lane >> 4;
  const int row = rt * 16 + m;            // < 2048 always
  const int col = ct * 16 + m;            // < 112 always (padded B)

  const v2f* Ap = (const v2f*)(X + (size_t)row * 64);  // A pairs along K
  const v2f* Bp = (const v2f*)MxP;                     // [32][112] pairs

  v8f acc = {0.f,0.f,0.f,0.f,0.f,0.f,0.f,0.f};
#pragma unroll
  for (int k0 = 0; k0 < 64; k0 += 4) {
    const int k2 = (k0 >> 1) + hi;        // pair index for this lane half
    v2f a = Ap[k2];
    v2f b = Bp[k2 * 112 + col];
    acc = wmma_f32_16x16x4(a, b, acc);
  }
  if (col < 103) {
#pragma unroll
    for (int v = 0; v < 8; ++v) {
      int orow = rt * 16 + v + hi * 8;
      outX[(size_t)orow * 103 + col] = acc[v];
    }
  }
}

// ---------------------------------------------------------------------------
// y: per image, T = Gr^T(44x32, pad 48) @ Yimg(32x32) in LDS, then T @ Gc.
// 6 waves per block (3 row tiles x 2 col tiles), 2048 blocks.
// Image staged once into LDS in pair-packed form -> b64 DS loads for B.
// ---------------------------------------------------------------------------
__global__ void y_resize_gemm(const float* __restrict__ Y,
                              const float* __restrict__ GrP,
                              const float* __restrict__ GcP,
                              float* __restrict__ outY) {
  __shared__ __align__(16) float Ypk[16 * 32 * 2];  // [k2][col] pairs, 4 KB
  __shared__ __align__(16) float Tsh[48][32];       // 6 KB

  const int img  = blockIdx.x;
  const int wave = threadIdx.x >> 5;
  const int lane = threadIdx.x & 31;
  const int m = lane & 15, hi = lane >> 4;
  const int ti = wave >> 1, tj = wave & 1;
  const float* Yb = Y + (size_t)img * 1024;

  // Stage image into LDS, pair-packed over K (row) dimension.
  for (int p = threadIdx.x; p < 512; p += blockDim.x) {
    int k2 = p >> 5, c = p & 31;
    Ypk[2 * p]     = Yb[k2 * 64 + c];        // row 2*k2
    Ypk[2 * p + 1] = Yb[k2 * 64 + 32 + c];   // row 2*k2+1
  }
  __syncthreads();

  // Stage 1: T tile (ti,tj) = Gr^T @ Yimg. A from GrP (rows >= 44 are zero).
  const int arow = ti * 16 + m;               // < 48 always (padded A)
  const int bcol = tj * 16 + m;
  const v2f* ApG = (const v2f*)GrP;           // [16][48] pairs
  const v2f* BpY = (const v2f*)Ypk;           // [16][32] pairs
  v8f acc = {0.f,0.f,0.f,0.f,0.f,0.f,0.f,0.f};
#pragma unroll
  for (int k0 = 0; k0 < 32; k0 += 4) {
    const int k2 = (k0 >> 1) + hi;
    v2f a = ApG[k2 * 48 + arow];
    v2f b = BpY[k2 * 32 + bcol];
    acc = wmma_f32_16x16x4(a, b, acc);
  }
#pragma unroll
  for (int v = 0; v < 8; ++v)
    Tsh[ti * 16 + v + hi * 8][tj * 16 + m] = acc[v];
  __syncthreads();

  // Stage 2: Out tile (ti,tj) = T(48x32) @ Gc(32x30 padded to 32).
  const v2f* ApT = (const v2f*)(&Tsh[ti * 16 + m][0]);  // contiguous K-pairs
  const v2f* BpG = (const v2f*)GcP;                     // [16][32] pairs
  v8f acc2 = {0.f,0.f,0.f,0.f,0.f,0.f,0.f,0.f};
#pragma unroll
  for (int k0 = 0; k0 < 32; k0 += 4) {
    const int k2 = (k0 >> 1) + hi;
    v2f a = ApT[k2];
    v2f b = BpG[k2 * 32 + bcol];
    acc2 = wmma_f32_16x16x4(a, b, acc2);
  }
  if (bcol < 30) {
#pragma unroll
    for (int v = 0; v < 8; ++v) {
      int orow = ti * 16 + v + hi * 8;
      if (orow < 44)
        outY[(size_t)img * 1320 + orow * 30 + bcol] = acc2[v];
    }
  }
}

// ---------------------------------------------------------------------------
// w: nearest with explicit scale factor, scale = f32(1/2.976744), clamp to 23.
// ---------------------------------------------------------------------------
__global__ void w_resize_nearest(const float* __restrict__ W,
                                 float* __restrict__ outW, int total) {
  int t = blockIdx.x * blockDim.x + threadIdx.x;
  if (t >= total) return;
  const float s = (float)(1.0 / 2.976744);
  int j = t % 71;
  int rest = t / 71;
  int i = rest % 71;
  int b = rest / 71;
  int sj = imin((int)floorf((float)j * s), 23);
  int si = imin((int)floorf((float)i * s), 23);
  outW[t] = W[(size_t)b * 576 + si * 24 + sj];
}

// ---------------------------------------------------------------------------
extern "C" void kernel_launch(void* const* d_in, const int* in_sizes, int n_in,
                              void* d_out, int out_size, void* d_ws, size_t ws_size,
                              hipStream_t stream) {
  const float* X = (const float*)d_in[0];   // (8,256,64)
  const float* Y = (const float*)d_in[1];   // (8,256,32,32)
  const float* W = (const float*)d_in[2];   // (8,256,24,24)

  float* out  = (float*)d_out;
  float* outX = out;                              // 2048*103
  float* outY = outX + 2048 * 103;                // 2048*44*30
  float* outW = outY + 2048 * 44 * 30;            // 2048*71*71

  // Workspace: builder ping-pong scratch, then the packed operand matrices.
  float* ws      = (float*)d_ws;
  float* scratch = ws;                                    // 3*2*64*192 = 73728 floats
  float* MxP     = scratch + 3 * 2 * BROWS * BPITCH;      // 32*112*2 = 7168 floats
  float* GrP     = MxP + 32 * 112 * 2;                    // 16*48*2  = 1536 floats
  float* GcP     = GrP + 16 * 48 * 2;                     // 16*32*2  = 1024 floats

  build_mats_kernel<<<3, 256, 0, stream>>>(scratch, MxP, GrP, GcP);
  x_resize_gemm<<<128 * 7, 32, 0, stream>>>(X, MxP, outX);
  y_resize_gemm<<<2048, 192, 0, stream>>>(Y, GrP, GcP, outY);
  const int totalW = 2048 * 71 * 71;
  w_resize_nearest<<<(totalW + 255) / 256, 256, 0, stream>>>(W, outW, totalW);
}